// NeuronBasedLanguageModel_41884521071079
// MI455X (gfx1250) — compile-verified
//
#include <hip/hip_runtime.h>
#include <hip/hip_bf16.h>
#include <math.h>

// ---- model dims (fixed by the reference) ----
constexpr int B  = 2;
constexpr int S  = 512;
constexpr int D  = 768;
constexpr int FF = 3072;
constexpr int H  = 12;
constexpr int L  = 4;
constexpr int V  = 32000;
constexpr int K  = 256;
constexpr int HD = D / H;          // 64
constexpr int N  = B * S;          // 1024 tokens

typedef __attribute__((ext_vector_type(2))) float v2f;
typedef __attribute__((ext_vector_type(8))) float v8f;

__device__ __forceinline__ float gelu_f(float z) {
    // exact GELU: 0.5*z*(1+erf(z/sqrt(2)))
    return 0.5f * z * (1.0f + erff(z * 0.70710678118654752f));
}

__device__ __forceinline__ v8f wmma_f32(v2f a, v2f b, v8f c) {
    return __builtin_amdgcn_wmma_f32_16x16x4_f32(false, a, false, b, (short)0, c, false, false);
}

// ---------------------------------------------------------------------------
// Embedding: x[n,d] = tok_emb[ids[n], d] + pos_emb[n % S, d]
// ---------------------------------------------------------------------------
__global__ void embed_kernel(const int* __restrict__ ids,
                             const float* __restrict__ tok,
                             const float* __restrict__ pos,
                             float* __restrict__ x) {
    int i = blockIdx.x * blockDim.x + threadIdx.x;
    if (i >= N * D) return;
    int n = i / D, d = i - n * D;
    int s = n % S;
    x[i] = tok[(size_t)ids[n] * D + d] + pos[(size_t)s * D + d];
}

// ---------------------------------------------------------------------------
// LayerNorm over last dim (row per block, 256 threads)
// ---------------------------------------------------------------------------
__global__ void __launch_bounds__(256)
ln_kernel(const float* __restrict__ in, float* __restrict__ out,
          const float* __restrict__ g, const float* __restrict__ b) {
    __shared__ float red[256];
    int row = blockIdx.x;
    const float* xr = in + (size_t)row * D;

    float s = 0.f;
    for (int d = threadIdx.x; d < D; d += 256) s += xr[d];
    red[threadIdx.x] = s;
    __syncthreads();
    for (int off = 128; off > 0; off >>= 1) {
        if (threadIdx.x < off) red[threadIdx.x] += red[threadIdx.x + off];
        __syncthreads();
    }
    float mean = red[0] * (1.0f / D);
    __syncthreads();

    float vs = 0.f;
    for (int d = threadIdx.x; d < D; d += 256) {
        float t = xr[d] - mean;
        vs += t * t;
    }
    red[threadIdx.x] = vs;
    __syncthreads();
    for (int off = 128; off > 0; off >>= 1) {
        if (threadIdx.x < off) red[threadIdx.x] += red[threadIdx.x + off];
        __syncthreads();
    }
    float rstd = rsqrtf(red[0] * (1.0f / D) + 1e-5f);

    for (int d = threadIdx.x; d < D; d += 256)
        out[(size_t)row * D + d] = (xr[d] - mean) * rstd * g[d] + b[d];
}

// ---------------------------------------------------------------------------
// WMMA fp32 GEMM:  C[n,m] = act( sum_k A[n,k]*W[m,k] + bias[m] ) + res[n,m]
//   A: [Ntok x Kd] row-major, W: [M x Kd] row-major  (C = A * W^T)
// Block = 128 threads = 4 waves (2x2). Block tile 64x128; wave tile 32x64:
// 2x4 register blocking -> 8 independent v_wmma_f32_16x16x4_f32 chains per
// K-step fed by 6 fragment loads. Fragments are software-pipelined (next
// K-step loaded before current WMMAs issue) so waits are partial and loads
// overlap the matrix pipe. Weights are L2-resident (192MB L2 >> 150MB total).
// ---------------------------------------------------------------------------
__device__ __forceinline__ void store_tile(v8f acc, int row0, int col0, int lane,
                                           const float* bias, const float* res,
                                           float* C, int M, int act) {
    int n  = lane & 15;           // column within tile
    int mb = (lane >> 4) * 8;     // row base within tile
    int col = col0 + n;
    float bval = bias ? bias[col] : 0.0f;
    #pragma unroll
    for (int r = 0; r < 8; ++r) {
        size_t off = (size_t)(row0 + mb + r) * M + col;
        float v = acc[r] + bval;
        if (act) v = gelu_f(v);
        if (res) v += res[off];
        C[off] = v;
    }
}

__global__ void __launch_bounds__(128)
gemm_wmma_nt(const float* __restrict__ A,
             const float* __restrict__ W,
             const float* __restrict__ bias,   // [M] or nullptr
             const float* __restrict__ res,    // [Ntok x M] or nullptr
             float* __restrict__ C,            // [Ntok x M]
             int Kd, int M, int act) {
    int wv   = threadIdx.x >> 5;
    int lane = threadIdx.x & 31;
    int row0 = blockIdx.y * 64  + (wv >> 1) * 32;   // two 16-row tiles per wave
    int col0 = blockIdx.x * 128 + (wv & 1) * 64;    // four 16-col tiles per wave

    // A 16x4 fp32 layout: lanes 0-15 -> M=lane, K={0,1}; lanes 16-31 -> M=lane-16, K={2,3}
    // B 4x16 fp32 layout (mirror): lanes 0-15 -> N=lane, K={0,1}; lanes 16-31 -> N=lane-16, K={2,3}
    int mr = lane & 15;
    int kh = (lane >> 4) * 2;
    const float* ap0 = A + (size_t)(row0 + mr) * Kd + kh;
    const float* ap1 = ap0 + (size_t)16 * Kd;
    const float* bp0 = W + (size_t)(col0 + mr) * Kd + kh;
    const float* bp1 = bp0 + (size_t)16 * Kd;
    const float* bp2 = bp0 + (size_t)32 * Kd;
    const float* bp3 = bp0 + (size_t)48 * Kd;

    v8f c00 = {}, c01 = {}, c02 = {}, c03 = {};
    v8f c10 = {}, c11 = {}, c12 = {}, c13 = {};

    // prologue: fragments for k = 0
    v2f a0 = *(const v2f*)(ap0);
    v2f a1 = *(const v2f*)(ap1);
    v2f b0 = *(const v2f*)(bp0);
    v2f b1 = *(const v2f*)(bp1);
    v2f b2 = *(const v2f*)(bp2);
    v2f b3 = *(const v2f*)(bp3);

    for (int k = 4; k < Kd; k += 4) {
        // prefetch next K-step while current WMMAs execute
        v2f a0n = *(const v2f*)(ap0 + k);
        v2f a1n = *(const v2f*)(ap1 + k);
        v2f b0n = *(const v2f*)(bp0 + k);
        v2f b1n = *(const v2f*)(bp1 + k);
        v2f b2n = *(const v2f*)(bp2 + k);
        v2f b3n = *(const v2f*)(bp3 + k);

        c00 = wmma_f32(a0, b0, c00);
        c01 = wmma_f32(a0, b1, c01);
        c02 = wmma_f32(a0, b2, c02);
        c03 = wmma_f32(a0, b3, c03);
        c10 = wmma_f32(a1, b0, c10);
        c11 = wmma_f32(a1, b1, c11);
        c12 = wmma_f32(a1, b2, c12);
        c13 = wmma_f32(a1, b3, c13);

        a0 = a0n; a1 = a1n;
        b0 = b0n; b1 = b1n; b2 = b2n; b3 = b3n;
    }
    // epilogue: last K-step
    c00 = wmma_f32(a0, b0, c00);
    c01 = wmma_f32(a0, b1, c01);
    c02 = wmma_f32(a0, b2, c02);
    c03 = wmma_f32(a0, b3, c03);
    c10 = wmma_f32(a1, b0, c10);
    c11 = wmma_f32(a1, b1, c11);
    c12 = wmma_f32(a1, b2, c12);
    c13 = wmma_f32(a1, b3, c13);

    store_tile(c00, row0,      col0 +  0, lane, bias, res, C, M, act);
    store_tile(c01, row0,      col0 + 16, lane, bias, res, C, M, act);
    store_tile(c02, row0,      col0 + 32, lane, bias, res, C, M, act);
    store_tile(c03, row0,      col0 + 48, lane, bias, res, C, M, act);
    store_tile(c10, row0 + 16, col0 +  0, lane, bias, res, C, M, act);
    store_tile(c11, row0 + 16, col0 + 16, lane, bias, res, C, M, act);
    store_tile(c12, row0 + 16, col0 + 32, lane, bias, res, C, M, act);
    store_tile(c13, row0 + 16, col0 + 48, lane, bias, res, C, M, act);
}

// ---------------------------------------------------------------------------
// Attention: one block per (b, h, q). qkv = [N, 3D] with q|k|v column blocks.
// Scores -> softmax -> weighted V sum. S=512, HD=64.
// ---------------------------------------------------------------------------
__global__ void __launch_bounds__(128)
attn_kernel(const float* __restrict__ qkv, float* __restrict__ o) {
    __shared__ float qs[HD];
    __shared__ float sc[S];
    __shared__ float red[128];

    int idx = blockIdx.x;
    int q = idx % S;
    int h = (idx / S) % H;
    int b = idx / (S * H);
    int rowq = b * S + q;

    const float* qp = qkv + (size_t)rowq * (3 * D) + h * HD;
    if (threadIdx.x < HD) qs[threadIdx.x] = qp[threadIdx.x];
    __syncthreads();

    for (int kk = threadIdx.x; kk < S; kk += 128) {
        const float* kp = qkv + (size_t)(b * S + kk) * (3 * D) + D + h * HD;
        float s = 0.f;
        #pragma unroll 8
        for (int d = 0; d < HD; ++d) s += qs[d] * kp[d];
        sc[kk] = s * 0.125f;   // HD^-0.5 = 1/8
    }
    __syncthreads();

    float m = -1e30f;
    for (int kk = threadIdx.x; kk < S; kk += 128) m = fmaxf(m, sc[kk]);
    red[threadIdx.x] = m;
    __syncthreads();
    for (int off = 64; off > 0; off >>= 1) {
        if (threadIdx.x < off)
            red[threadIdx.x] = fmaxf(red[threadIdx.x], red[threadIdx.x + off]);
        __syncthreads();
    }
    m = red[0];
    __syncthreads();

    float ssum = 0.f;
    for (int kk = threadIdx.x; kk < S; kk += 128) {
        float e = expf(sc[kk] - m);
        sc[kk] = e;
        ssum += e;
    }
    red[threadIdx.x] = ssum;
    __syncthreads();
    for (int off = 64; off > 0; off >>= 1) {
        if (threadIdx.x < off) red[threadIdx.x] += red[threadIdx.x + off];
        __syncthreads();
    }
    float inv = 1.0f / red[0];

    for (int d = threadIdx.x; d < HD; d += 128) {
        float acc = 0.f;
        for (int kk = 0; kk < S; ++kk)
            acc += sc[kk] * qkv[(size_t)(b * S + kk) * (3 * D) + 2 * D + h * HD + d];
        o[(size_t)rowq * D + h * HD + d] = acc * inv;
    }
}

// ---------------------------------------------------------------------------
// Top-K by exact rank (total order: value desc, index asc). rank<K selected,
// written at slot==rank -> deterministic and matches jax.lax.top_k ordering.
// ---------------------------------------------------------------------------
__global__ void __launch_bounds__(256)
topk_kernel(const float* __restrict__ scores, int* __restrict__ idx) {
    __shared__ float s[FF];
    int row = blockIdx.x;
    for (int j = threadIdx.x; j < FF; j += 256)
        s[j] = scores[(size_t)row * FF + j];
    __syncthreads();
    for (int e = threadIdx.x; e < FF; e += 256) {
        float v = s[e];
        int rank = 0;
        for (int j = 0; j < FF; ++j) {
            float w = s[j];
            rank += (w > v) || ((w == v) && (j < e));
        }
        if (rank < K)
            idx[(size_t)row * K + rank] = e;
    }
}

// ---------------------------------------------------------------------------
// Gathered FFN: z[k]=dot(h2[n], W1[idx[k]]); a=gelu(z);
//               x[n,d] += sum_k a[k]*W2[d, idx[k]]
// One block (256 threads == K) per token. Gathered rows are L2-resident.
// ---------------------------------------------------------------------------
__global__ void __launch_bounds__(256)
ffn_gather_kernel(const float* __restrict__ h2, const int* __restrict__ idx,
                  const float* __restrict__ W1, const float* __restrict__ W2,
                  float* __restrict__ x) {
    __shared__ float hs[D];
    __shared__ float a[K];
    __shared__ int   si[K];
    int n = blockIdx.x;

    for (int d = threadIdx.x; d < D; d += 256)
        hs[d] = h2[(size_t)n * D + d];
    int e = idx[(size_t)n * K + threadIdx.x];
    si[threadIdx.x] = e;
    __syncthreads();

    const float* w1 = W1 + (size_t)e * D;
    float z = 0.f;
    #pragma unroll 4
    for (int d = 0; d < D; ++d) z += hs[d] * w1[d];
    a[threadIdx.x] = gelu_f(z);
    __syncthreads();

    for (int d = threadIdx.x; d < D; d += 256) {
        float acc = 0.f;
        #pragma unroll 4
        for (int k = 0; k < K; ++k)
            acc += a[k] * W2[(size_t)d * FF + si[k]];
        x[(size_t)n * D + d] += acc;
    }
}

// ---------------------------------------------------------------------------
// Host orchestration
// ---------------------------------------------------------------------------
extern "C" void kernel_launch(void* const* d_in, const int* in_sizes, int n_in,
                              void* d_out, int out_size, void* d_ws, size_t ws_size,
                              hipStream_t stream) {
    const int*   input_ids  = (const int*)  d_in[0];
    const float* tok_emb    = (const float*)d_in[1];
    const float* pos_emb    = (const float*)d_in[2];
    const float* in_proj_w  = (const float*)d_in[3];
    const float* in_proj_b  = (const float*)d_in[4];
    const float* attn_out_w = (const float*)d_in[5];
    const float* attn_out_b = (const float*)d_in[6];
    const float* n1_g       = (const float*)d_in[7];
    const float* n1_b       = (const float*)d_in[8];
    const float* n2_g       = (const float*)d_in[9];
    const float* n2_b       = (const float*)d_in[10];
    const float* W1         = (const float*)d_in[11];
    const float* W2         = (const float*)d_in[12];
    const float* Wr1        = (const float*)d_in[13];
    const float* Wr2        = (const float*)d_in[14];
    const float* rn_g       = (const float*)d_in[15];
    const float* rn_b       = (const float*)d_in[16];
    const float* fn_g       = (const float*)d_in[17];
    const float* fn_b       = (const float*)d_in[18];
    const float* outp_w     = (const float*)d_in[19];
    const float* outp_b     = (const float*)d_in[20];
    float* logits = (float*)d_out;

    // workspace carve-out (~41 MB of floats; every element written before read)
    float* ws     = (float*)d_ws;
    float* x      = ws;                         // [N, D]
    float* h      = x      + (size_t)N * D;     // [N, D]
    float* qkv    = h      + (size_t)N * D;     // [N, 3D]
    float* o      = qkv    + (size_t)N * 3 * D; // [N, D]
    float* h2     = o      + (size_t)N * D;     // [N, D]
    float* xr     = h2     + (size_t)N * D;     // [N, D]
    float* hr     = xr     + (size_t)N * D;     // [N, D]
    float* scores = hr     + (size_t)N * D;     // [N, FF]
    int*   idx    = (int*)(scores + (size_t)N * FF); // [N, K]

    embed_kernel<<<(N * D + 255) / 256, 256, 0, stream>>>(input_ids, tok_emb, pos_emb, x);

    for (int l = 0; l < L; ++l) {
        // --- attention block ---
        ln_kernel<<<N, 256, 0, stream>>>(x, h, n1_g + (size_t)l * D, n1_b + (size_t)l * D);
        gemm_wmma_nt<<<dim3(3 * D / 128, N / 64), 128, 0, stream>>>(
            h, in_proj_w + (size_t)l * 3 * D * D, in_proj_b + (size_t)l * 3 * D,
            nullptr, qkv, D, 3 * D, 0);
        attn_kernel<<<B * H * S, 128, 0, stream>>>(qkv, o);
        gemm_wmma_nt<<<dim3(D / 128, N / 64), 128, 0, stream>>>(
            o, attn_out_w + (size_t)l * D * D, attn_out_b + (size_t)l * D,
            x, x, D, D, 0);                       // x += o @ Wout^T + b

        // --- dynamic FFN ---
        ln_kernel<<<N, 256, 0, stream>>>(x, h2, n2_g + (size_t)l * D, n2_b + (size_t)l * D);
        ln_kernel<<<N, 256, 0, stream>>>(h2, xr, rn_g + (size_t)l * D, rn_b + (size_t)l * D);
        gemm_wmma_nt<<<dim3(D / 128, N / 64), 128, 0, stream>>>(
            xr, Wr1 + (size_t)l * D * D, nullptr, nullptr, hr, D, D, 1);   // gelu
        gemm_wmma_nt<<<dim3(FF / 128, N / 64), 128, 0, stream>>>(
            hr, Wr2 + (size_t)l * FF * D, nullptr, nullptr, scores, D, FF, 0);
        topk_kernel<<<N, 256, 0, stream>>>(scores, idx);
        ffn_gather_kernel<<<N, 256, 0, stream>>>(
            h2, idx, W1 + (size_t)l * FF * D, W2 + (size_t)l * D * FF, x);
    }

    // --- final LN + logits ---
    ln_kernel<<<N, 256, 0, stream>>>(x, h, fn_g, fn_b);
    gemm_wmma_nt<<<dim3(V / 128, N / 64), 128, 0, stream>>>(
        h, outp_w, outp_b, nullptr, logits, D, V, 0);
}